// Embedding2ScoreSAN_3135326126725
// MI455X (gfx1250) — compile-verified
//
#include <hip/hip_runtime.h>
#include <stdint.h>

typedef __attribute__((ext_vector_type(2))) float v2f;
typedef __attribute__((ext_vector_type(8))) float v8f;
typedef __attribute__((ext_vector_type(4))) unsigned int u32x4;
typedef __attribute__((ext_vector_type(4))) int i32x4;
typedef __attribute__((ext_vector_type(8))) int i32x8;

#define HDIM   256
#define BATCHN 1024
#define NNODE  32
#define VOCAB  50000

// GEMM tiling
#define BM 128      // block rows (8 waves x 16)
#define BN 64       // block cols (4 wmma subtiles per wave)
#define KC 32       // K chunk staged per TDM transfer
#define SA 34       // LDS row stride: KC + 2 pad dwords (from TDM pad feature)

// ---------------------------------------------------------------------------
// TDM: issue a 2D tile load (tile_rows x tile_cols fp32) global -> LDS.
// D# built per CDNA5 ISA ch.8:
//   group0: count=1 | lds_addr | global_addr(57b) | type=2
//   group1: data_size=4B, pad_enable, pad_interval=4 (32 dwords),
//           pad_amount=1 (2 dwords) -> LDS row stride = 34 floats,
//           tensor_dim0/1 = remaining extents (OOB rows/cols zero-filled),
//           tensor_dim0_stride = row stride in elements.
//   groups2/3: zero (2D tensor).
// ---------------------------------------------------------------------------
__device__ __forceinline__ void tdm_load_tile(uint32_t lds_off, const float* g,
                                              int rows_rem, int cols_rem,
                                              int row_stride_elems,
                                              int tile_rows, int tile_cols) {
  const uint64_t ga = (uint64_t)(uintptr_t)g;
  u32x4 g0 = { 1u,                                    // count=1 (valid user D#)
               lds_off,                               // lds_addr (bytes)
               (uint32_t)ga,                          // global_addr[31:0]
               (uint32_t)(ga >> 32) | (2u << 30) };   // global_addr[56:32] | type=2
  const uint32_t d0 = (uint32_t)cols_rem;             // tensor_dim0 (elements)
  const uint32_t d1 = (uint32_t)rows_rem;             // tensor_dim1 (rows)
  i32x8 g1 = { (int)((2u << 16)        // data_size = 4 bytes
                    | (1u << 20)       // pad_enable
                    | (4u << 22)       // pad_interval: every 32 dwords
                    | (1u << 25)),     // pad_amount: 2 dwords
               (int)((d0 & 0xFFFFu) << 16),
               (int)((d0 >> 16) | ((d1 & 0xFFFFu) << 16)),
               (int)((d1 >> 16) | ((uint32_t)tile_cols << 16)),   // tile_dim0
               (int)(uint32_t)tile_rows,                          // tile_dim1
               (int)(uint32_t)row_stride_elems,                   // dim0_stride lo
               0, 0 };
  i32x4 gz = {0, 0, 0, 0};
#if __clang_major__ >= 23
  i32x8 gz8 = {0, 0, 0, 0, 0, 0, 0, 0};
  __builtin_amdgcn_tensor_load_to_lds(g0, g1, gz, gz, gz8, 0);
#else
  __builtin_amdgcn_tensor_load_to_lds(g0, g1, gz, gz, 0);
#endif
}

// ---------------------------------------------------------------------------
// Kernel 1: v_n for the last step of each session -> both halves of xcat.
// ---------------------------------------------------------------------------
__global__ void vn_kernel(const float* __restrict__ node_emb,
                          const int*   __restrict__ seq,
                          const float* __restrict__ ilen,
                          const int*   __restrict__ slen,
                          float*       __restrict__ xcat) {
  const int b = blockIdx.x;
  const int t = threadIdx.x;
  int off = 0;
  for (int i = 0; i < b; ++i) off += slen[i];
  const int last = off + slen[b] - 1;

  float acc = 0.f;
#pragma unroll
  for (int g = 0; g < 3; ++g) {
    const int s = seq[last * 3 + g];
    if (s >= 0 && s < NNODE)          // s == NNODE is the zero pad row
      acc += node_emb[((long)b * NNODE + s) * HDIM + t];
  }
  const float val = acc / ilen[last];
  xcat[(long)b * (2 * HDIM) + t] = val;
  xcat[(long)b * (2 * HDIM) + HDIM + t] = val;
}

// ---------------------------------------------------------------------------
// fp32 WMMA GEMM with TDM double-buffered LDS staging:
//   C[M x N] = A[M x K (lda)] * W[N x K]^T  (+bias)(relu)(+res)
// Block: 256 threads (8 waves) -> 128x64 tile; wave w owns rows [w*16, w*16+16).
// Fragment layout for V_WMMA_F32_16X16X4_F32 (ISA 7.12.2):
//   lanes 0-15 -> K=0,1 ; lanes 16-31 -> K=2,3 ; C: VGPR r -> rows r / r+8.
// ---------------------------------------------------------------------------
__global__ void __launch_bounds__(256)
gemm_wt_wmma(const float* __restrict__ A, int lda,
             const float* __restrict__ W,
             const float* __restrict__ bias,
             const float* __restrict__ res, int ldres,
             float* __restrict__ C, int ldc,
             int M, int N, int K, int relu) {
  __shared__ __align__(16) float sA[2][BM * SA];
  __shared__ __align__(16) float sW[2][BN * SA];

  const int lane = threadIdx.x & 31;
  const int wave = threadIdx.x >> 5;
  const int m0 = blockIdx.y * BM;
  const int n0 = blockIdx.x * BN;
  const int NC = K / KC;

  if (wave == 0) {   // wave 0 drives the TDM pipeline (TENSORcnt is per-wave)
    tdm_load_tile((uint32_t)(uintptr_t)&sA[0][0], A + (long)m0 * lda,
                  M - m0, K, lda, BM, KC);
    tdm_load_tile((uint32_t)(uintptr_t)&sW[0][0], W + (long)n0 * K,
                  N - n0, K, K, BN, KC);
  }

  const int r    = lane & 15;          // row within A-subtile / col within B,C
  const int koff = (lane >> 4) << 1;   // 0 (lanes 0-15) or 2 (lanes 16-31)
  v8f acc[4] = {};

  for (int c = 0; c < NC; ++c) {
    const int cur = c & 1;
    if (wave == 0) {
      if (c + 1 < NC) {
        const int k0 = (c + 1) * KC;
        tdm_load_tile((uint32_t)(uintptr_t)&sA[cur ^ 1][0],
                      A + (long)m0 * lda + k0, M - m0, K - k0, lda, BM, KC);
        tdm_load_tile((uint32_t)(uintptr_t)&sW[cur ^ 1][0],
                      W + (long)n0 * K + k0, N - n0, K - k0, K, BN, KC);
        __builtin_amdgcn_s_wait_tensorcnt(2);   // in-order: chunk c resident
      } else {
        __builtin_amdgcn_s_wait_tensorcnt(0);
      }
    }
    __syncthreads();                            // publish LDS tile to all waves

    const float* Ap = &sA[cur][(wave * 16 + r) * SA + koff];
    const float* Wp = &sW[cur][r * SA + koff];
    for (int k = 0; k < KC; k += 4) {
      const v2f a = *(const v2f*)(Ap + k);
#pragma unroll
      for (int j = 0; j < 4; ++j) {
        const v2f b = *(const v2f*)(Wp + j * (16 * SA) + k);
        acc[j] = __builtin_amdgcn_wmma_f32_16x16x4_f32(
            false, a, false, b, (short)0, acc[j], false, false);
      }
    }
    __syncthreads();                            // done reading before overwrite
  }

  // Epilogue: bias / relu / residual, guarded for ragged N (N % 16 == 0).
  const int mbase = m0 + wave * 16 + ((lane >> 4) << 3);
#pragma unroll
  for (int j = 0; j < 4; ++j) {
    if (n0 + j * 16 >= N) continue;             // wave-uniform guard
    const int n = n0 + j * 16 + r;
    const float bb = bias ? bias[n] : 0.f;
#pragma unroll
    for (int rr = 0; rr < 8; ++rr) {
      float v = acc[j][rr] + bb;
      if (relu) v = fmaxf(v, 0.f);
      if (res)  v += res[(long)(mbase + rr) * ldres + n];
      C[(long)(mbase + rr) * ldc + n] = v;
    }
  }
}

// ---------------------------------------------------------------------------
// y_hat[b] = dot(s_h[b], emb_weight[cue[b]]) -- one wave per session
// ---------------------------------------------------------------------------
__global__ void yhat_kernel(const float* __restrict__ s_h,
                            const float* __restrict__ emb,
                            const int*   __restrict__ cue,
                            float*       __restrict__ out) {
  const int b = blockIdx.x * (blockDim.x >> 5) + (threadIdx.x >> 5);
  const int lane = threadIdx.x & 31;
  if (b >= BATCHN) return;
  const float* sv = s_h + (long)b * HDIM;
  const float* ev = emb + (long)cue[b] * HDIM;
  float acc = 0.f;
  for (int i = lane; i < HDIM; i += 32) acc += sv[i] * ev[i];
  for (int o = 16; o > 0; o >>= 1) acc += __shfl_xor(acc, o, 32);
  if (lane == 0) out[b] = acc;
}

// ---------------------------------------------------------------------------
static inline void launch_gemm(const float* A, int lda, const float* W,
                               const float* bias, const float* res, int ldres,
                               float* C, int ldc, int M, int N, int K,
                               int relu, hipStream_t stream) {
  dim3 grid((N + BN - 1) / BN, M / BM);
  gemm_wt_wmma<<<grid, 256, 0, stream>>>(A, lda, W, bias, res, ldres,
                                         C, ldc, M, N, K, relu);
}

extern "C" void kernel_launch(void* const* d_in, const int* in_sizes, int n_in,
                              void* d_out, int out_size, void* d_ws, size_t ws_size,
                              hipStream_t stream) {
  (void)in_sizes; (void)n_in; (void)out_size; (void)ws_size;
  const float* node_embedding = (const float*)d_in[0];
  const float* emb_weight     = (const float*)d_in[1];
  const float* itemset_len    = (const float*)d_in[2];
  const float* in_proj_w      = (const float*)d_in[3];
  const float* in_proj_b      = (const float*)d_in[4];
  const float* out_proj_w     = (const float*)d_in[5];
  const float* out_proj_b     = (const float*)d_in[6];
  const float* d1_w           = (const float*)d_in[7];
  const float* d1_b           = (const float*)d_in[8];
  const float* d2_w           = (const float*)d_in[9];
  const float* d2_b           = (const float*)d_in[10];
  const float* W3_w           = (const float*)d_in[11];
  const float* W3_b           = (const float*)d_in[12];
  // d_in[13] = batch (sections are uniformly 32; unused)
  const int*   sequence       = (const int*)d_in[14];
  const int*   sequence_len   = (const int*)d_in[15];
  const int*   cue            = (const int*)d_in[16];

  float* ws   = (float*)d_ws;
  float* xcat = ws;                                   // 1024 x 512
  float* t1   = xcat + (long)BATCHN * 2 * HDIM;       // 1024 x 256
  float* attn = t1   + (long)BATCHN * HDIM;           // 1024 x 256
  float* t2   = attn + (long)BATCHN * HDIM;           // 1024 x 256
  float* s_h  = t2   + (long)BATCHN * HDIM;           // 1024 x 256

  float* yhat   = (float*)d_out;
  float* scores = (float*)d_out + BATCHN;

  // 1) v_n -> both halves of xcat
  vn_kernel<<<BATCHN, HDIM, 0, stream>>>(node_embedding, sequence,
                                         itemset_len, sequence_len, xcat);

  // 2) 3 SAN blocks (x lives in xcat[:,256:512], lda/ldc = 512)
  const float* Wv = in_proj_w + 2 * HDIM * HDIM;   // in_proj_w[2H:]
  const float* bv = in_proj_b + 2 * HDIM;
  for (int it = 0; it < 3; ++it) {
    launch_gemm(xcat + HDIM, 2 * HDIM, Wv,         bv,         nullptr, 0,
                t1,   HDIM, BATCHN, HDIM, HDIM, 0, stream);
    launch_gemm(t1,   HDIM,  out_proj_w, out_proj_b, nullptr, 0,
                attn, HDIM, BATCHN, HDIM, HDIM, 0, stream);
    launch_gemm(attn, HDIM,  d1_w,       d1_b,       nullptr, 0,
                t2,   HDIM, BATCHN, HDIM, HDIM, 1, stream);          // relu
    launch_gemm(t2,   HDIM,  d2_w,       d2_b,       attn, HDIM,     // +attn
                xcat + HDIM, 2 * HDIM, BATCHN, HDIM, HDIM, 0, stream);
  }

  // 3) s_h = concat(v_n, s_g) @ W3^T + b3   (K = 512)
  launch_gemm(xcat, 2 * HDIM, W3_w, W3_b, nullptr, 0,
              s_h, HDIM, BATCHN, HDIM, 2 * HDIM, 0, stream);

  // 4) y_hat
  yhat_kernel<<<BATCHN / 8, 256, 0, stream>>>(s_h, emb_weight, cue, yhat);

  // 5) all_scores = s_h @ emb_weight^T (HBM-bound: 205MB out; emb stays in L2)
  launch_gemm(s_h, HDIM, emb_weight, nullptr, nullptr, 0,
              scores, VOCAB, BATCHN, VOCAB, HDIM, 0, stream);
}